// InstanceGrouping_68135361183938
// MI455X (gfx1250) — compile-verified
//
#include <hip/hip_runtime.h>

#define K_SEL 2000
#define HH 75
#define WW 100
#define NPIX 7500
#define BATCH 8
#define CDIM 512
#define FDIM 128
#define NBR_MAX 12
#define MROWS (BATCH * K_SEL)  // 16000 contiguous rows across batches

typedef __attribute__((ext_vector_type(2))) float v2f;
typedef __attribute__((ext_vector_type(8))) float v8f;

__device__ __forceinline__ unsigned tokey(float f) {
    unsigned u = __float_as_uint(f);
    return (u & 0x80000000u) ? ~u : (u | 0x80000000u);
}

// ---------------- Top-K select (exact, per-batch radix select + ordered compaction) --------
__device__ __forceinline__ unsigned block_incl_scan(unsigned v, unsigned* sbuf) {
    const int tid = threadIdx.x;
    sbuf[tid] = v;
    __syncthreads();
    for (int off = 1; off < 512; off <<= 1) {
        unsigned t = (tid >= off) ? sbuf[tid - off] : 0u;
        __syncthreads();
        sbuf[tid] += t;
        __syncthreads();
    }
    return sbuf[tid];
}

__global__ __launch_bounds__(512) void topk_kernel(const float* __restrict__ pred,
                                                   int* __restrict__ sidx,
                                                   int* __restrict__ out_idx) {
    const int b = blockIdx.x;
    const int tid = threadIdx.x;
    const float* sc = pred + (size_t)b * 3 * NPIX;  // channel 0 plane

    __shared__ unsigned hist[256];
    __shared__ unsigned sbuf[512];
    __shared__ unsigned sh_prefix, sh_need, sh_sbase, sh_ebase;
    if (tid == 0) { sh_prefix = 0u; sh_need = K_SEL; sh_sbase = 0u; sh_ebase = 0u; }

    // 4-pass MSB-first radix select of the K-th largest key
    for (int p = 3; p >= 0; --p) {
        if (tid < 256) hist[tid] = 0u;
        __syncthreads();
        unsigned pref = sh_prefix;
        for (int i = tid; i < NPIX; i += 512) {
            unsigned u = tokey(sc[i]);
            bool ok;
            if (p < 3) ok = ((u >> (8 * p + 8)) == (pref >> (8 * p + 8)));
            else ok = true;
            if (ok) atomicAdd(&hist[(u >> (8 * p)) & 255u], 1u);
        }
        __syncthreads();
        if (tid == 0) {
            unsigned need = sh_need, cum = 0u;
            int d = 255;
            for (; d > 0; --d) {
                unsigned h = hist[d];
                if (cum + h >= need) break;
                cum += h;
            }
            sh_need = need - cum;
            sh_prefix = pref | ((unsigned)d << (8 * p));
        }
        __syncthreads();
    }
    const unsigned thresh = sh_prefix;
    const unsigned need_eq = sh_need;

    // ordered compaction: select key>thresh, plus first `need_eq` keys==thresh (lowest indices)
    for (int base = 0; base < NPIX; base += 512) {
        __syncthreads();
        unsigned ebase = sh_ebase;
        unsigned sb = sh_sbase;
        int i = base + tid;
        bool valid = i < NPIX;
        unsigned u = valid ? tokey(sc[i]) : 0u;
        unsigned eq = (valid && u == thresh) ? 1u : 0u;
        unsigned gt = (valid && u > thresh) ? 1u : 0u;
        __syncthreads();
        unsigned e_incl = block_incl_scan(eq, sbuf);
        unsigned e_excl = e_incl - eq;
        unsigned e_tot = sbuf[511];
        unsigned sel = (gt || (eq && (ebase + e_excl) < need_eq)) ? 1u : 0u;
        __syncthreads();
        unsigned s_incl = block_incl_scan(sel, sbuf);
        unsigned s_excl = s_incl - sel;
        unsigned s_tot = sbuf[511];
        if (sel) {
            unsigned pos = sb + s_excl;
            if (pos < K_SEL) {
                sidx[b * K_SEL + pos] = i;
                out_idx[b * K_SEL + pos] = i;
            }
        }
        __syncthreads();
        if (tid == 0) { sh_ebase = ebase + e_tot; sh_sbase = sb + s_tot; }
    }
}

// ---------------- Neighbor lists: adjacency has <=9 nnz/row within flat window +/-101 -------
__global__ __launch_bounds__(256) void neighbors_kernel(const int* __restrict__ sidx,
                                                        int* __restrict__ ncnt,
                                                        int* __restrict__ nbr) {
    int t = blockIdx.x * blockDim.x + threadIdx.x;
    if (t >= BATCH * K_SEL) return;
    int b = t / K_SEL;
    const int* s = sidx + b * K_SEL;
    int idx = s[t % K_SEL];
    int py = idx / WW, px = idx % WW;
    int lo = idx - (WW + 1);
    int L = 0, R = K_SEL;
    while (L < R) { int m = (L + R) >> 1; if (s[m] < lo) L = m + 1; else R = m; }
    int cnt = 0;
    for (int j = L; j < K_SEL && cnt < NBR_MAX; ++j) {
        int q = s[j];
        if (q > idx + (WW + 1)) break;
        int qy = q / WW, qx = q % WW;
        int dy = qy - py; if (dy < 0) dy = -dy;
        int dx = qx - px; if (dx < 0) dx = -dx;
        if (dy <= 1 && dx <= 1) nbr[t * NBR_MAX + cnt++] = j;
    }
    ncnt[t] = cnt;
}

// ---------------- t1[b,i,:512] = sum_{j in N(i)} clip(feature[b,:,idx_j%75,idx_j/75],0,6) ---
__global__ __launch_bounds__(512) void spmm1_kernel(const float* __restrict__ feat,
                                                    const int* __restrict__ sidx,
                                                    const int* __restrict__ ncnt,
                                                    const int* __restrict__ nbr,
                                                    float* __restrict__ t1) {
    const int b = blockIdx.y, i = blockIdx.x;
    const int t = b * K_SEL + i;
    __shared__ int cnt_s;
    __shared__ int off_s[NBR_MAX];
    if (threadIdx.x == 0) cnt_s = ncnt[t];
    __syncthreads();
    if ((int)threadIdx.x < cnt_s) {
        int j = nbr[t * NBR_MAX + threadIdx.x];
        int q = sidx[b * K_SEL + j];
        off_s[threadIdx.x] = (q % HH) * WW + (q / HH);  // reference's transposed gather
    }
    __syncthreads();
    const int c = threadIdx.x;
    const float* fp = feat + ((size_t)b * CDIM + c) * NPIX;
    float acc = 0.f;
    for (int n = 0; n < cnt_s; ++n) {
        if (n + 1 < cnt_s) __builtin_prefetch(fp + off_s[n + 1], 0, 0);  // global_prefetch_b8
        float v = fp[off_s[n]];
        acc += fminf(fmaxf(v, 0.f), 6.f);
    }
    t1[(size_t)t * CDIM + c] = acc;
}

// ---------------- GEMM1: x1 = t1(16000x512) @ w1(512x128) + b1[row%2000] -------------------
// One wave computes a 64x16 strip: 4 M-tiles share each B fragment (1.5 VMEM per WMMA).
__global__ __launch_bounds__(256) void gemm1_kernel(const float* __restrict__ t1,
                                                    const float* __restrict__ w1,
                                                    const float* __restrict__ b1,
                                                    float* __restrict__ x1) {
    const int wid = threadIdx.x >> 5;
    const int lane = threadIdx.x & 31;
    const int job = blockIdx.x * 8 + wid;  // 250 blocks * 8 waves = 2000 jobs
    const int m0 = (job >> 3) * 64;        // 250 M-groups of 64 rows
    const int n0 = (job & 7) * 16;         // 8 N-tiles
    const int half = lane >> 4, lm = lane & 15;

    const float* A0 = t1 + ((size_t)(m0 + lm)) * CDIM + 2 * half;
    const float* B = w1 + (size_t)(2 * half) * FDIM + n0 + lm;
    v8f acc[4] = {};
#pragma unroll 2
    for (int k = 0; k < CDIM; k += 4) {
        v2f bb;
        bb.x = B[(size_t)k * FDIM];
        bb.y = B[(size_t)(k + 1) * FDIM];
        v2f a[4];
#pragma unroll
        for (int t4 = 0; t4 < 4; ++t4)
            a[t4] = *reinterpret_cast<const v2f*>(A0 + (size_t)t4 * 16 * CDIM + k);
#pragma unroll
        for (int t4 = 0; t4 < 4; ++t4)
            acc[t4] = __builtin_amdgcn_wmma_f32_16x16x4_f32(false, a[t4], false, bb,
                                                            (short)0, acc[t4], false, false);
    }
#pragma unroll
    for (int t4 = 0; t4 < 4; ++t4) {
        // 2000 % 16 == 0 and tiles are 16-aligned => a 16-row tile never straddles a batch
        // boundary: row % 2000 == ((m0 + t4*16) % 2000) + rr + 8*half  (scalar modulo).
        const int tb = (m0 + t4 * 16) % K_SEL;
#pragma unroll
        for (int rr = 0; rr < 8; ++rr) {
            int row = m0 + t4 * 16 + rr + 8 * half;
            int f = n0 + lm;
            x1[(size_t)row * FDIM + f] =
                acc[t4][rr] + b1[(size_t)(tb + rr + 8 * half) * FDIM + f];
        }
    }
}

// ---------------- t2[b,i,:128] = sum_{j in N(i)} x1[b,j,:128] ------------------------------
__global__ __launch_bounds__(128) void spmm2_kernel(const float* __restrict__ x1,
                                                    const int* __restrict__ ncnt,
                                                    const int* __restrict__ nbr,
                                                    float* __restrict__ t2) {
    const int b = blockIdx.y, i = blockIdx.x;
    const int t = b * K_SEL + i;
    __shared__ int cnt_s;
    __shared__ int j_s[NBR_MAX];
    if (threadIdx.x == 0) cnt_s = ncnt[t];
    __syncthreads();
    if ((int)threadIdx.x < cnt_s) j_s[threadIdx.x] = nbr[t * NBR_MAX + threadIdx.x];
    __syncthreads();
    const int f = threadIdx.x;
    float acc = 0.f;
    for (int n = 0; n < cnt_s; ++n)
        acc += x1[((size_t)b * K_SEL + j_s[n]) * FDIM + f];
    t2[(size_t)t * FDIM + f] = acc;
}

// ---------------- GEMM2 + epilogue: out = tanh(t2 @ w2 + b2[row%2000] + x1) ----------------
__global__ __launch_bounds__(256) void gemm2_kernel(const float* __restrict__ t2,
                                                    const float* __restrict__ w2,
                                                    const float* __restrict__ b2,
                                                    const float* __restrict__ x1,
                                                    float* __restrict__ out) {
    const int wid = threadIdx.x >> 5;
    const int lane = threadIdx.x & 31;
    const int job = blockIdx.x * 8 + wid;
    const int m0 = (job >> 3) * 64;
    const int n0 = (job & 7) * 16;
    const int half = lane >> 4, lm = lane & 15;

    const float* A0 = t2 + ((size_t)(m0 + lm)) * FDIM + 2 * half;
    const float* B = w2 + (size_t)(2 * half) * FDIM + n0 + lm;
    v8f acc[4] = {};
#pragma unroll 2
    for (int k = 0; k < FDIM; k += 4) {
        v2f bb;
        bb.x = B[(size_t)k * FDIM];
        bb.y = B[(size_t)(k + 1) * FDIM];
        v2f a[4];
#pragma unroll
        for (int t4 = 0; t4 < 4; ++t4)
            a[t4] = *reinterpret_cast<const v2f*>(A0 + (size_t)t4 * 16 * FDIM + k);
#pragma unroll
        for (int t4 = 0; t4 < 4; ++t4)
            acc[t4] = __builtin_amdgcn_wmma_f32_16x16x4_f32(false, a[t4], false, bb,
                                                            (short)0, acc[t4], false, false);
    }
#pragma unroll
    for (int t4 = 0; t4 < 4; ++t4) {
        const int tb = (m0 + t4 * 16) % K_SEL;  // see gemm1 comment
#pragma unroll
        for (int rr = 0; rr < 8; ++rr) {
            int row = m0 + t4 * 16 + rr + 8 * half;
            int f = n0 + lm;
            size_t o = (size_t)row * FDIM + f;
            float v = acc[t4][rr] + b2[(size_t)(tb + rr + 8 * half) * FDIM + f] + x1[o];
            out[o] = tanhf(v);
        }
    }
}

extern "C" void kernel_launch(void* const* d_in, const int* in_sizes, int n_in,
                              void* d_out, int out_size, void* d_ws, size_t ws_size,
                              hipStream_t stream) {
    const float* feat = (const float*)d_in[0];  // (8,512,75,100)
    const float* pred = (const float*)d_in[1];  // (8,3,75,100)
    const float* w1 = (const float*)d_in[2];    // (512,128)
    const float* b1 = (const float*)d_in[3];    // (2000,128)
    const float* w2 = (const float*)d_in[4];    // (128,128)
    const float* b2 = (const float*)d_in[5];    // (2000,128)

    float* outf = (float*)d_out;                                 // (8,2000,128) floats
    int* out_idx = (int*)(outf + (size_t)MROWS * FDIM);          // (8,2000) int32

    float* t1 = (float*)d_ws;                                    // 16000*512
    float* x1 = t1 + (size_t)MROWS * CDIM;                       // 16000*128
    float* t2 = x1 + (size_t)MROWS * FDIM;                       // 16000*128
    int* sidx = (int*)(t2 + (size_t)MROWS * FDIM);               // 16000
    int* ncnt = sidx + MROWS;                                    // 16000
    int* nbr = ncnt + MROWS;                                     // 16000*12

    topk_kernel<<<BATCH, 512, 0, stream>>>(pred, sidx, out_idx);
    neighbors_kernel<<<(MROWS + 255) / 256, 256, 0, stream>>>(sidx, ncnt, nbr);
    spmm1_kernel<<<dim3(K_SEL, BATCH), 512, 0, stream>>>(feat, sidx, ncnt, nbr, t1);
    gemm1_kernel<<<250, 256, 0, stream>>>(t1, w1, b1, x1);
    spmm2_kernel<<<dim3(K_SEL, BATCH), 128, 0, stream>>>(x1, ncnt, nbr, t2);
    gemm2_kernel<<<250, 256, 0, stream>>>(t2, w2, b2, x1, outf);
}